// AccumulatorCell_41790031790408
// MI455X (gfx1250) — compile-verified
//
#include <hip/hip_runtime.h>

// Problem constants (from the reference)
#define T_DIM   512
#define UNITS   180
#define HALFV   90.0f
#define KC      32      // K-chunk (time steps staged per iteration)
#define NWAVES  6
#define BLOCK   (NWAVES * 32)
#define MPAD    192     // padded column (m) dimension: 12 WMMA tiles
#define ROWH    96      // i-rows handled per workgroup (2 halves cover 192 >= 180)

typedef __attribute__((ext_vector_type(2))) float v2f;
typedef __attribute__((ext_vector_type(8))) float v8f;

// out[b, i*180 + j] = sum_t s[b,t]*valid[b,t]*e0(loc0,i+1)*e1(loc1,(i+j)%180+1)
// Computed as C = A^T B per batch via FP32 WMMA, then column-rotated store.
__global__ __launch_bounds__(BLOCK) void rbf_accum_wmma(
    const float* __restrict__ inputs,  // [B, T, 6] : 4 signal ch + 2 loc
    const float* __restrict__ a0p,
    const float* __restrict__ a1p,
    float* __restrict__ out)           // [B, 180*180]
{
    const int b    = blockIdx.x;
    const int hb   = blockIdx.y;        // row half: i in [hb*96, hb*96+96)
    const int tid  = threadIdx.x;
    const int wave = tid >> 5;
    const int lane = tid & 31;
    const int hl   = lane >> 4;         // lane half (0: lanes 0-15, 1: 16-31)
    const int lm   = lane & 15;

    __shared__ float sA[KC * ROWH];     // A-chunk: [k][i_local], 12 KB
    __shared__ float sB[KC * MPAD];     // B-chunk: [k][m],       24 KB
    __shared__ float sW[KC], sL0[KC], sL1[KC];

    const float a0 = a0p[0];
    const float a1 = a1p[0];

    v8f acc[12];
#pragma unroll
    for (int jt = 0; jt < 12; ++jt)
#pragma unroll
        for (int p = 0; p < 8; ++p) acc[jt][p] = 0.0f;

    const float* inB = inputs + (size_t)b * T_DIM * 6;

    for (int tc = 0; tc < T_DIM; tc += KC) {
        // ---- stage per-timestep scalars ----
        if (tid < KC) {
            const float* x = inB + (size_t)(tc + tid) * 6;
            float s0 = x[0];
            float l0 = x[4];
            float l1 = x[5];
            float valid = (l0 > 0.0f && l1 > 0.0f) ? 1.0f : 0.0f;
            sW[tid]  = s0 * valid;
            sL0[tid] = l0;
            sL1[tid] = l1;
        }
        __syncthreads();

        // ---- build A-chunk: A[k][ci] = s*valid * exp(-a0*d0^2), i = hb*96+ci
        for (int e = tid; e < KC * ROWH; e += BLOCK) {
            int t  = e / ROWH;
            int ci = e - t * ROWH;
            int i  = hb * ROWH + ci;
            float v = 0.0f;
            if (i < UNITS) {
                float c = (float)(i + 1);
                float d = 1.0f - fabsf(fabsf(sL0[t] - c) * (1.0f / HALFV) - 1.0f);
                v = sW[t] * __expf(-a0 * d * d);
            }
            sA[e] = v;
        }
        // ---- build B-chunk: B[k][m] = exp(-a1*d1^2)
        for (int e = tid; e < KC * MPAD; e += BLOCK) {
            int t = e / MPAD;
            int m = e - t * MPAD;
            float v = 0.0f;
            if (m < UNITS) {
                float c = (float)(m + 1);
                float d = 1.0f - fabsf(fabsf(sL1[t] - c) * (1.0f / HALFV) - 1.0f);
                v = __expf(-a1 * d * d);
            }
            sB[e] = v;
        }
        __syncthreads();

        // ---- WMMA: C[i,m] += sum_k A[k][i] * B[k][m]  (FP32 16x16x4) ----
        // A-operand layout: vgpr r, lane: element (M = lane%16, K = r + 2*(lane/16))
        // B-operand layout: vgpr r, lane: element (K = r + 2*(lane/16), N = lane%16)
        const int ciBase = wave * 16;
#pragma unroll
        for (int k0 = 0; k0 < KC; k0 += 4) {
            const int krow = k0 + 2 * hl;
            v2f x;
            x[0] = sA[(krow + 0) * ROWH + ciBase + lm];
            x[1] = sA[(krow + 1) * ROWH + ciBase + lm];
#pragma unroll
            for (int jt = 0; jt < 12; ++jt) {
                v2f y;
                y[0] = sB[(krow + 0) * MPAD + jt * 16 + lm];
                y[1] = sB[(krow + 1) * MPAD + jt * 16 + lm];
                acc[jt] = __builtin_amdgcn_wmma_f32_16x16x4_f32(
                    false, x, false, y, (short)0, acc[jt], false, false);
            }
        }
        __syncthreads();
    }

    // ---- epilogue: out[b, i*180 + j] = C[i, m],  j = (m - i) mod 180 ----
    const int i0 = hb * ROWH + wave * 16;
    float* outB = out + (size_t)b * (UNITS * UNITS);
#pragma unroll
    for (int jt = 0; jt < 12; ++jt) {
#pragma unroll
        for (int p = 0; p < 8; ++p) {
            int i = i0 + p + 8 * hl;   // C/D layout: vgpr p, M = p + 8*(lane/16)
            int m = jt * 16 + lm;
            if (i < UNITS && m < UNITS) {
                int j = m - i;
                if (j < 0) j += UNITS;
                outB[i * UNITS + j] = acc[jt][p];
            }
        }
    }
}

extern "C" void kernel_launch(void* const* d_in, const int* in_sizes, int n_in,
                              void* d_out, int out_size, void* d_ws, size_t ws_size,
                              hipStream_t stream) {
    const float* inputs = (const float*)d_in[0];
    const float* a0     = (const float*)d_in[1];
    const float* a1     = (const float*)d_in[2];
    float* out          = (float*)d_out;

    const int B = in_sizes[0] / (T_DIM * 6);   // 128
    dim3 grid(B, 2, 1);
    rbf_accum_wmma<<<grid, BLOCK, 0, stream>>>(inputs, a0, a1, out);
}